// HeteroGNN_18416819765449
// MI455X (gfx1250) — compile-verified
//
#include <hip/hip_runtime.h>

#define NPER 200000
#define NSK  20000
#define NAG  50000
#define NE   1000000

typedef __attribute__((ext_vector_type(2))) float v2f;
typedef __attribute__((ext_vector_type(8))) float v8f;

// ---------------------------------------------------------------------------
// Edge scatter: agg[dst] += x_src[src], cnt[dst] += 1  (mean aggregation)
// One thread per (edge, feature). agg for the largest dst type is 102MB and
// L2-resident (192MB), so f32 global atomics are cheap.
// ---------------------------------------------------------------------------
__global__ void scatter_kernel(const float* __restrict__ xsrc,
                               const int* __restrict__ src,
                               const int* __restrict__ dst,
                               float* __restrict__ agg,
                               float* __restrict__ cnt,
                               int nE, int ld) {
  long long t = (long long)blockIdx.x * blockDim.x + threadIdx.x;
  if (t >= ((long long)nE << ld)) return;
  int e = (int)(t >> ld);
  int c = (int)(t & ((1 << ld) - 1));
  int s = src[e];
  int d = dst[e];
  atomicAdd(agg + (((long long)d) << ld) + c, xsrc[(((long long)s) << ld) + c]);
  if (c == 0) atomicAdd(cnt + d, 1.0f);
}

// ---------------------------------------------------------------------------
// f32 WMMA GEMM: C[n x 128] (=|+=) (A/cnt)[n x K] @ (B1(+B2))[K x 128] (+bias)
// 256 threads = 8 waves; block covers 128 rows, each wave a 16x128 strip via
// 8 x V_WMMA_F32_16X16X4_F32 accumulators. Weights staged in 64KB LDS.
// A frag (16x4): lane l -> row l&15, K = k0 + 2*(l>>4) + {0,1}
// B frag (4x16): lane l -> col l&15, K = k0 + 2*(l>>4) + {0,1}
// C frag: VGPR i -> row i + 8*(l>>4), col l&15
// ---------------------------------------------------------------------------
__global__ void gemm_wmma_kernel(const float* __restrict__ A,
                                 const float* __restrict__ cnt,   // null => scale 1
                                 const float* __restrict__ B1,
                                 const float* __restrict__ B2,    // optional (root: Wr1+Wr2)
                                 const float* __restrict__ bias1, // optional
                                 const float* __restrict__ bias2, // optional
                                 float* __restrict__ C,
                                 int n, int K, int accum, int relu) {
  __shared__ float Bs[128 * 128];
  int tid = threadIdx.x;
  for (int i = tid; i < K * 128; i += 256) {
    float v = B1[i];
    if (B2) v += B2[i];
    Bs[i] = v;
  }
  __syncthreads();

  int wave = tid >> 5, lane = tid & 31;
  int hi = lane >> 4, l15 = lane & 15;
  int tile = blockIdx.x * 128 + wave * 16;
  if (tile >= n) return;

  int rowA = tile + l15; if (rowA > n - 1) rowA = n - 1;
  float scale = 1.0f;
  if (cnt) scale = 1.0f / fmaxf(cnt[rowA], 1.0f);

  v8f acc[8];
  if (accum) {
#pragma unroll
    for (int nt = 0; nt < 8; ++nt)
#pragma unroll
      for (int i = 0; i < 8; ++i) {
        int r = tile + i + 8 * hi; if (r > n - 1) r = n - 1;
        acc[nt][i] = C[(long long)r * 128 + nt * 16 + l15];
      }
  } else {
#pragma unroll
    for (int nt = 0; nt < 8; ++nt) {
      v8f z = {0.f, 0.f, 0.f, 0.f, 0.f, 0.f, 0.f, 0.f};
      acc[nt] = z;
    }
  }

  const float* Arow = A + (long long)rowA * K;
  for (int k0 = 0; k0 < K; k0 += 4) {
    float2 av = *(const float2*)(Arow + k0 + 2 * hi);
    v2f a; a.x = av.x * scale; a.y = av.y * scale;
#pragma unroll
    for (int nt = 0; nt < 8; ++nt) {
      v2f b;
      b.x = Bs[(k0 + 2 * hi) * 128 + nt * 16 + l15];
      b.y = Bs[(k0 + 2 * hi + 1) * 128 + nt * 16 + l15];
      acc[nt] = __builtin_amdgcn_wmma_f32_16x16x4_f32(
          false, a, false, b, (short)0, acc[nt], false, false);
    }
  }

#pragma unroll
  for (int nt = 0; nt < 8; ++nt) {
    float bv = 0.f;
    if (bias1) bv += bias1[nt * 16 + l15];
    if (bias2) bv += bias2[nt * 16 + l15];
#pragma unroll
    for (int i = 0; i < 8; ++i) {
      int r = tile + i + 8 * hi;
      if (r < n) {
        float v = acc[nt][i] + bv;
        if (relu) v = fmaxf(v, 0.f);
        C[(long long)r * 128 + nt * 16 + l15] = v;
      }
    }
  }
}

// ---------------------------------------------------------------------------
// Host-side orchestration
// ---------------------------------------------------------------------------
struct RelP { const float *Wl, *bl, *Wr; };
enum { HS = 0, RHS = 1, OS_ = 2, ROS = 3, LP_ = 4, RLP = 5 };

static inline void run_root(const float* x, int K, const RelP& r1, const RelP& r2,
                            float* out, int n, hipStream_t s) {
  gemm_wmma_kernel<<<dim3((n + 127) / 128), dim3(256), 0, s>>>(
      x, nullptr, r1.Wr, r2.Wr, r1.bl, r2.bl, out, n, K, 0, 0);
}

static inline void run_rel(const float* xsrc, int d, const int* esrc, const int* edst,
                           int n_dst, const float* Wl, float* out, int relu,
                           float* agg, float* cnt, hipStream_t s) {
  hipMemsetAsync(agg, 0, (size_t)n_dst * d * sizeof(float), s);
  hipMemsetAsync(cnt, 0, (size_t)n_dst * sizeof(float), s);
  int ld = (d == 128) ? 7 : 6;
  long long total = (long long)NE << ld;
  scatter_kernel<<<dim3((unsigned)((total + 255) / 256)), dim3(256), 0, s>>>(
      xsrc, esrc, edst, agg, cnt, NE, ld);
  gemm_wmma_kernel<<<dim3((n_dst + 127) / 128), dim3(256), 0, s>>>(
      agg, cnt, Wl, nullptr, nullptr, nullptr, out, n_dst, d, 1, relu);
}

static void run_layer(const float* xP, const float* xS, const float* xA,
                      int dP, int dS, int dA, const RelP* P,
                      const int* const* e,  // e[2*rel], e[2*rel+1]
                      float* oP, float* oS, float* oA,
                      float* agg, float* cnt, hipStream_t s) {
  // skill = sage(person->skill) + sage(agent->skill)
  run_root(xS, dS, P[HS], P[OS_], oS, NSK, s);
  run_rel(xP, dP, e[2 * HS], e[2 * HS + 1], NSK, P[HS].Wl, oS, 0, agg, cnt, s);
  run_rel(xA, dA, e[2 * OS_], e[2 * OS_ + 1], NSK, P[OS_].Wl, oS, 1, agg, cnt, s);
  // person = sage(skill->person) + sage(agent->person)
  run_root(xP, dP, P[RHS], P[RLP], oP, NPER, s);
  run_rel(xS, dS, e[2 * RHS], e[2 * RHS + 1], NPER, P[RHS].Wl, oP, 0, agg, cnt, s);
  run_rel(xA, dA, e[2 * RLP], e[2 * RLP + 1], NPER, P[RLP].Wl, oP, 1, agg, cnt, s);
  // agent = sage(skill->agent) + sage(person->agent)
  run_root(xA, dA, P[ROS], P[LP_], oA, NAG, s);
  run_rel(xS, dS, e[2 * ROS], e[2 * ROS + 1], NAG, P[ROS].Wl, oA, 0, agg, cnt, s);
  run_rel(xP, dP, e[2 * LP_], e[2 * LP_ + 1], NAG, P[LP_].Wl, oA, 1, agg, cnt, s);
}

extern "C" void kernel_launch(void* const* d_in, const int* in_sizes, int n_in,
                              void* d_out, int out_size, void* d_ws, size_t ws_size,
                              hipStream_t stream) {
  const float* xP = (const float*)d_in[0];
  const float* xS = (const float*)d_in[1];
  const float* xA = (const float*)d_in[2];
  const int* e[12];
  for (int i = 0; i < 12; ++i) e[i] = (const int*)d_in[3 + i];

  RelP L1[6], L2[6];
  const int base = 15;
  if (n_in > base + 1 && in_sizes[base + 1] == 128) {
    // dict-insertion order: rel {hs,rhs,os,ros,lp,rlp}, fields (Wl, bl, Wr)
    static const int ord[6] = {HS, RHS, OS_, ROS, LP_, RLP};
    for (int L = 0; L < 2; ++L) {
      RelP* P = L ? L2 : L1;
      for (int j = 0; j < 6; ++j) {
        int b = base + L * 18 + j * 3;
        P[ord[j]].Wl = (const float*)d_in[b + 0];
        P[ord[j]].bl = (const float*)d_in[b + 1];
        P[ord[j]].Wr = (const float*)d_in[b + 2];
      }
    }
  } else {
    // sorted-key order: rel {hs,lp,os,rhs,rlp,ros}, fields (Wl, Wr, bl)
    static const int ord[6] = {HS, LP_, OS_, RHS, RLP, ROS};
    for (int L = 0; L < 2; ++L) {
      RelP* P = L ? L2 : L1;
      for (int j = 0; j < 6; ++j) {
        int b = base + L * 18 + j * 3;
        P[ord[j]].Wl = (const float*)d_in[b + 0];
        P[ord[j]].Wr = (const float*)d_in[b + 1];
        P[ord[j]].bl = (const float*)d_in[b + 2];
      }
    }
  }

  // Workspace layout (floats): agg[NPER*128] | cnt[NPER] | xp1 | xs1 | xa1
  float* ws  = (float*)d_ws;
  float* agg = ws;
  float* cnt = agg + (size_t)NPER * 128;
  float* xp1 = cnt + NPER;
  float* xs1 = xp1 + (size_t)NPER * 128;
  float* xa1 = xs1 + (size_t)NSK * 128;

  float* outP = (float*)d_out;
  float* outS = outP + (size_t)NPER * 128;
  float* outA = outS + (size_t)NSK * 128;

  // Layer 1 (input dims 128/64/128), ReLU fused into last accumulate per type
  run_layer(xP, xS, xA, 128, 64, 128, L1, e, xp1, xs1, xa1, agg, cnt, stream);
  // Layer 2 (all dims 128), outputs straight into d_out, final ReLU fused
  run_layer(xp1, xs1, xa1, 128, 128, 128, L2, e, outP, outS, outA, agg, cnt, stream);

  (void)out_size; (void)ws_size;
}